// CoAttention_58583353917660
// MI455X (gfx1250) — compile-verified
//
#include <hip/hip_runtime.h>
#include <hip/hip_bf16.h>

#define B_  4
#define LC_ 512
#define LW_ 256
#define H_  512

typedef __attribute__((ext_vector_type(2))) float v2f;
typedef __attribute__((ext_vector_type(8))) float v8f;

// ---------------------------------------------------------------------------
// Fast tanh: CDNA5 has native V_TANH_F32 (a TRANS op that co-executes with
// VALU). Fall back to exp2/rcp trans ops if the builtin is absent.
// ---------------------------------------------------------------------------
__device__ __forceinline__ float fast_tanh(float x) {
#if __has_builtin(__builtin_amdgcn_tanhf)
    return __builtin_amdgcn_tanhf(x);
#else
    // tanh(x) = 1 - 2/(e^{2x}+1);  e^{2x} = exp2(x * 2*log2(e))
    float e = __builtin_amdgcn_exp2f(x * 2.8853900817779268f);
    return 1.0f - 2.0f * __builtin_amdgcn_rcpf(e + 1.0f);
#endif
}

// ---------------------------------------------------------------------------
// One 16x16 f32 WMMA tile, K-loop in steps of 4 (V_WMMA_F32_16X16X4_F32).
// ISA layout (32-bit A 16x4): per-lane k = vgpr + 2*(lane>>4), m = lane&15.
// B mirrors with n = lane&15.  C/D: vgpr r, group g: (m = r + 8g, n = lane&15).
// ---------------------------------------------------------------------------
__device__ __forceinline__ v8f wmma_f32_gemm_tile(
    const float* __restrict__ A, long sAm, long sAk,
    const float* __restrict__ Bp, long sBk, long sBn,
    int K, int lane)
{
    int mn = lane & 15;
    int g2 = (lane >> 4) << 1;
    const float* arow = A + (long)mn * sAm;
    const float* bcol = Bp + (long)mn * sBn;
    v8f acc = {};
    for (int k0 = 0; k0 < K; k0 += 4) {
        long ka = (long)(k0 + g2);
        v2f a, b;
        a.x = arow[ka * sAk];
        a.y = arow[(ka + 1) * sAk];
        b.x = bcol[ka * sBk];
        b.y = bcol[(ka + 1) * sBk];
        acc = __builtin_amdgcn_wmma_f32_16x16x4_f32(
            false, a, false, b, (short)0, acc, false, false);
    }
    return acc;
}

// ---------------------------------------------------------------------------
// Y[M,N] = X[M,K] @ W[N,K]^T + bias   (torch Linear). One wave per 16x16 tile.
// ---------------------------------------------------------------------------
__global__ __launch_bounds__(256) void proj_gemm_kernel(
    const float* __restrict__ X, const float* __restrict__ W,
    const float* __restrict__ bias, float* __restrict__ Y,
    int M, int N, int K)
{
    int wave = (blockIdx.x * blockDim.x + threadIdx.x) >> 5;
    int lane = threadIdx.x & 31;
    int tilesN = N >> 4;
    int tilesTotal = (M >> 4) * tilesN;
    if (wave >= tilesTotal) return;
    int tm = wave / tilesN;
    int tn = wave - tm * tilesN;

    const float* A  = X + (long)tm * 16 * K;      // A[m][k], row-major
    const float* Bp = W + (long)tn * 16 * K;      // B[k][n] = W[n*K + k]
    v8f acc = wmma_f32_gemm_tile(A, K, 1, Bp, 1, K, K, lane);

    int n = (tn << 4) + (lane & 15);
    int g = lane >> 4;
    float bv = bias[n];
#pragma unroll
    for (int r = 0; r < 8; ++r) {
        int m = (tm << 4) + r + (g << 3);
        Y[(long)m * N + n] = acc[r] + bv;
    }
}

// ---------------------------------------------------------------------------
// Batched GEMM with arbitrary strides (covers A and A^T): Y[b] = A[b] @ B[b].
// ---------------------------------------------------------------------------
__global__ __launch_bounds__(256) void batched_gemm_kernel(
    const float* __restrict__ Abase, long strideAb, long sAm, long sAk,
    const float* __restrict__ Bbase, long strideBb, long sBk, long sBn,
    float* __restrict__ Ybase, long strideYb,
    int M, int N, int K)
{
    int b    = blockIdx.y;
    int wave = (blockIdx.x * blockDim.x + threadIdx.x) >> 5;
    int lane = threadIdx.x & 31;
    int tilesN = N >> 4;
    int tilesTotal = (M >> 4) * tilesN;
    if (wave >= tilesTotal) return;
    int tm = wave / tilesN;
    int tn = wave - tm * tilesN;

    const float* A  = Abase + (long)b * strideAb + (long)tm * 16 * sAm;
    const float* Bp = Bbase + (long)b * strideBb + (long)tn * 16 * sBn;
    v8f acc = wmma_f32_gemm_tile(A, sAm, sAk, Bp, sBk, sBn, K, lane);

    int n = (tn << 4) + (lane & 15);
    int g = lane >> 4;
    float* Y = Ybase + (long)b * strideYb;
#pragma unroll
    for (int r = 0; r < 8; ++r) {
        int m = (tm << 4) + r + (g << 3);
        Y[(long)m * N + n] = acc[r];
    }
}

// ---------------------------------------------------------------------------
// The hotspot: scores[b,c,w] = sum_h Wv[h]*tanh(cP[b,c,h]+wP[b,w,h]) + Wvb.
// 32x32 (c,w) tile per block, 64-wide h chunks staged in LDS, 4x2 register
// blocking per thread (128 threads/block). 512 blocks; per 4-h step:
// 7 ds_read_b128 feed 32 tanh(trans) + 64 VALU ops (co-execution).
// ---------------------------------------------------------------------------
#define TC_  32
#define TW_  32
#define HC_  64
#define LDST 68   // pad: bounded bank conflicts, keeps float4 alignment

__global__ __launch_bounds__(128) void scores_kernel(
    const float* __restrict__ cP, const float* __restrict__ wP,
    const float* __restrict__ Wv, const float* __restrict__ Wvb,
    float* __restrict__ scores)
{
    __shared__ float cT[TC_][LDST];
    __shared__ float wT[TW_][LDST];
    __shared__ float wv[HC_];

    int b   = blockIdx.z;
    int c0  = blockIdx.y * TC_;
    int w0  = blockIdx.x * TW_;
    int tid = threadIdx.x;
    int tx  = tid & 15;   // -> 2 w's  (16*2 = 32)
    int ty  = tid >> 4;   // -> 4 c's  (8*4  = 32)

    float acc[4][2] = {};

    for (int h0 = 0; h0 < H_; h0 += HC_) {
        __syncthreads();
        // stage 32x64 tiles of cP and wP (float4, coalesced): 512 slots each
#pragma unroll
        for (int i = 0; i < 4; ++i) {
            int idx = tid + (i << 7);
            int row = idx >> 4;
            int col = (idx & 15) << 2;
            *(float4*)&cT[row][col] =
                *(const float4*)&cP[((long)(b * LC_ + c0 + row)) * H_ + h0 + col];
            *(float4*)&wT[row][col] =
                *(const float4*)&wP[((long)(b * LW_ + w0 + row)) * H_ + h0 + col];
        }
        if (tid < (HC_ >> 2))
            *(float4*)&wv[tid << 2] = *(const float4*)&Wv[h0 + (tid << 2)];
        __syncthreads();

#pragma unroll 4
        for (int hh = 0; hh < HC_; hh += 4) {
            float4 vv = *(float4*)&wv[hh];
            float4 cv[4], wvv[2];
#pragma unroll
            for (int i = 0; i < 4; ++i)
                cv[i] = *(float4*)&cT[(ty << 2) + i][hh];
#pragma unroll
            for (int j = 0; j < 2; ++j)
                wvv[j] = *(float4*)&wT[(tx << 1) + j][hh];
#define ACC_STEP(q)                                               \
            {                                                     \
                _Pragma("unroll")                                 \
                for (int i = 0; i < 4; ++i) {                     \
                    _Pragma("unroll")                             \
                    for (int j = 0; j < 2; ++j)                   \
                        acc[i][j] += vv.q *                       \
                            fast_tanh(cv[i].q + wvv[j].q);        \
                }                                                 \
            }
            ACC_STEP(x) ACC_STEP(y) ACC_STEP(z) ACC_STEP(w)
#undef ACC_STEP
        }
    }

    float bv = Wvb[0];
    long base = ((long)(b * LC_ + c0 + (ty << 2))) * LW_ + w0 + (tx << 1);
#pragma unroll
    for (int i = 0; i < 4; ++i) {
#pragma unroll
        for (int j = 0; j < 2; ++j)
            scores[base + (long)i * LW_ + j] = acc[i][j] + bv;
    }
}

// ---------------------------------------------------------------------------
// Softmax over w (rows of 256): one wave32 per row, 8 elems/lane.
// ---------------------------------------------------------------------------
__global__ __launch_bounds__(256) void softmax_w_kernel(
    const float* __restrict__ scores, float* __restrict__ out)
{
    int wave = (blockIdx.x * blockDim.x + threadIdx.x) >> 5;
    int lane = threadIdx.x & 31;
    const float* row = scores + (long)wave * LW_;
    float v[8];
    float m = -3.4e38f;
#pragma unroll
    for (int i = 0; i < 8; ++i) {
        v[i] = row[lane + (i << 5)];
        m = fmaxf(m, v[i]);
    }
#pragma unroll
    for (int off = 16; off > 0; off >>= 1)
        m = fmaxf(m, __shfl_xor(m, off, 32));
    float s = 0.f;
#pragma unroll
    for (int i = 0; i < 8; ++i) {
        v[i] = __builtin_amdgcn_exp2f((v[i] - m) * 1.4426950408889634f);
        s += v[i];
    }
#pragma unroll
    for (int off = 16; off > 0; off >>= 1)
        s += __shfl_xor(s, off, 32);
    float inv = 1.0f / s;
    float* o = out + (long)wave * LW_;
#pragma unroll
    for (int i = 0; i < 8; ++i)
        o[lane + (i << 5)] = v[i] * inv;
}

// ---------------------------------------------------------------------------
// Softmax over c (columns of 512): block = 16 columns x 16 c-strided groups,
// LDS tree reduction -> 64 blocks instead of 4, 16x shorter load chains.
// Loads coalesced across the 16 consecutive w per group.
// ---------------------------------------------------------------------------
#define SMC_W 16
#define SMC_G 16
#define L2E_  1.4426950408889634f

__global__ __launch_bounds__(256) void softmax_c_kernel(
    const float* __restrict__ scores, float* __restrict__ out)
{
    __shared__ float red[SMC_G][SMC_W];
    int col0 = blockIdx.x * SMC_W;          // 0 .. B*LW-16
    int b  = col0 / LW_;
    int w0 = col0 - b * LW_;
    int wl = threadIdx.x & (SMC_W - 1);
    int cg = threadIdx.x >> 4;              // 0..15
    int w  = w0 + wl;

    const float* base = scores + (long)b * LC_ * LW_ + w;

    // pass 1: column max (strided partials, LDS reduce)
    float m = -3.4e38f;
    for (int c = cg; c < LC_; c += SMC_G)
        m = fmaxf(m, base[(long)c * LW_]);
    red[cg][wl] = m;
    __syncthreads();
    m = red[0][wl];
#pragma unroll
    for (int i = 1; i < SMC_G; ++i)
        m = fmaxf(m, red[i][wl]);
    __syncthreads();

    // pass 2: sum of exp
    float s = 0.f;
    for (int c = cg; c < LC_; c += SMC_G)
        s += __builtin_amdgcn_exp2f((base[(long)c * LW_] - m) * L2E_);
    red[cg][wl] = s;
    __syncthreads();
    s = 0.f;
#pragma unroll
    for (int i = 0; i < SMC_G; ++i)
        s += red[i][wl];
    float inv = 1.0f / s;

    // pass 3: write normalized
    float* obase = out + (long)b * LC_ * LW_ + w;
    for (int c = cg; c < LC_; c += SMC_G)
        obase[(long)c * LW_] =
            __builtin_amdgcn_exp2f((base[(long)c * LW_] - m) * L2E_) * inv;
}

// ---------------------------------------------------------------------------
extern "C" void kernel_launch(void* const* d_in, const int* in_sizes, int n_in,
                              void* d_out, int out_size, void* d_ws, size_t ws_size,
                              hipStream_t stream)
{
    const float* char_hidden = (const float*)d_in[0];
    const float* word_hidden = (const float*)d_in[1];
    const float* Wc_w = (const float*)d_in[2];
    const float* Wc_b = (const float*)d_in[3];
    const float* Ww_w = (const float*)d_in[4];
    const float* Ww_b = (const float*)d_in[5];
    const float* Wv_w = (const float*)d_in[6];
    const float* Wv_b = (const float*)d_in[7];

    float* ws       = (float*)d_ws;
    float* cP       = ws;                                   // B*LC*H
    float* wP       = cP + (size_t)B_ * LC_ * H_;           // B*LW*H
    float* scores   = wP + (size_t)B_ * LW_ * H_;           // B*LC*LW
    float* char_att = scores + (size_t)B_ * LC_ * LW_;      // B*LC*LW
    float* word_att = char_att + (size_t)B_ * LC_ * LW_;    // B*LC*LW

    float* char_context = (float*)d_out;                    // B*LC*H
    float* word_context = char_context + (size_t)B_ * LC_ * H_;  // B*LW*H

    // 1) cP = char_hidden @ Wc_w^T + Wc_b   (M=2048, N=512, K=512)
    {
        int M = B_ * LC_, N = H_, K = H_;
        int waves = (M >> 4) * (N >> 4);
        proj_gemm_kernel<<<(waves * 32 + 255) / 256, 256, 0, stream>>>(
            char_hidden, Wc_w, Wc_b, cP, M, N, K);
    }
    // 2) wP = word_hidden @ Ww_w^T + Ww_b   (M=1024, N=512, K=512)
    {
        int M = B_ * LW_, N = H_, K = H_;
        int waves = (M >> 4) * (N >> 4);
        proj_gemm_kernel<<<(waves * 32 + 255) / 256, 256, 0, stream>>>(
            word_hidden, Ww_w, Ww_b, wP, M, N, K);
    }
    // 3) scores (the tanh hotspot): 8 x 16 x 4 = 512 blocks of 128 threads
    {
        dim3 grid(LW_ / TW_, LC_ / TC_, B_);
        scores_kernel<<<grid, 128, 0, stream>>>(cP, wP, Wv_w, Wv_b, scores);
    }
    // 4) softmaxes
    softmax_w_kernel<<<(B_ * LC_ * 32) / 256, 256, 0, stream>>>(scores, char_att);
    softmax_c_kernel<<<(B_ * LW_) / SMC_W, 256, 0, stream>>>(scores, word_att);
    // 5) char_context[b] = char_att[b] (512x256) @ word_hidden[b] (256x512)
    {
        int M = LC_, N = H_, K = LW_;
        int waves = (M >> 4) * (N >> 4);
        dim3 grid((waves * 32 + 255) / 256, B_);
        batched_gemm_kernel<<<grid, 256, 0, stream>>>(
            char_att, (long)LC_ * LW_, LW_, 1,
            word_hidden, (long)LW_ * H_, H_, 1,
            char_context, (long)LC_ * H_, M, N, K);
    }
    // 6) word_context[b] = word_att[b]^T (256x512) @ char_hidden[b] (512x512)
    {
        int M = LW_, N = H_, K = LC_;
        int waves = (M >> 4) * (N >> 4);
        dim3 grid((waves * 32 + 255) / 256, B_);
        batched_gemm_kernel<<<grid, 256, 0, stream>>>(
            word_att, (long)LC_ * LW_, 1, LW_,
            char_hidden, (long)LC_ * H_, H_, 1,
            word_context, (long)LW_ * H_, M, N, K);
    }
}